// QuinLoss_6811818131844
// MI455X (gfx1250) — compile-verified
//
#include <hip/hip_runtime.h>
#include <hip/hip_bf16.h>

typedef __attribute__((ext_vector_type(2))) float v2f;
typedef __attribute__((ext_vector_type(4))) float v4f;
typedef __attribute__((ext_vector_type(8))) float v8f;

#define NN 8192
#define DD 128
#define NTILE (NN / 16)  // 512 column tiles

// ---------------------------------------------------------------------------
// Kernel 1: per-row squared norms. 8 waves/block, one wave per row.
// ---------------------------------------------------------------------------
__global__ __launch_bounds__(256) void sqnorm_kernel(const float* __restrict__ X,
                                                     float* __restrict__ sq) {
  const int wave = threadIdx.x >> 5;
  const int lane = threadIdx.x & 31;
  const int row = blockIdx.x * 8 + wave;
  v4f x = ((const v4f*)(X + (size_t)row * DD))[lane];
  float s = x.x * x.x + x.y * x.y + x.z * x.z + x.w * x.w;
#pragma unroll
  for (int off = 16; off > 0; off >>= 1) s += __shfl_xor(s, off, 32);
  if (lane == 0) sq[row] = s;
}

// ---------------------------------------------------------------------------
// Kernel 2: fused GEMM (X X^T via V_WMMA_F32_16X16X4_F32) + distance epilogue
// + per-row hardest-positive (argmax) / hardest-negative (argmin).
// Block = 16 rows, 8 waves stripe all 512 column tiles -> row reductions stay
// entirely inside the block (no global atomics). dist written once, NT hint.
// ---------------------------------------------------------------------------
__global__ __launch_bounds__(256) void dist_tile_kernel(
    const float* __restrict__ X, const long long* __restrict__ tgt,
    const float* __restrict__ sq, float* __restrict__ dist,
    float* __restrict__ dist_ap, float* __restrict__ dist_an,
    int* __restrict__ p2, int* __restrict__ n1) {
  const int wave = threadIdx.x >> 5;
  const int lane = threadIdx.x & 31;
  const int lh = lane & 15;   // position within half-wave
  const int hi = lane >> 4;   // which half-wave
  const int i0 = blockIdx.x * 16;

  __shared__ float lds_pv[8][16];
  __shared__ int lds_pj[8][16];
  __shared__ float lds_nv[8][16];
  __shared__ int lds_nj[8][16];

  // Preload all A fragments (16 rows x K=128) per the 16x4 f32 A layout:
  // lanes 0-15: M=lane, K = 4s+{0,1}; lanes 16-31: M=lane-16, K = 4s+{2,3}.
  const float* arow = X + (size_t)(i0 + lh) * DD + 2 * hi;
  v2f a[32];
#pragma unroll
  for (int s = 0; s < 32; ++s) a[s] = *(const v2f*)(arow + 4 * s);

  float sqi[8];
  long long ti[8];
#pragma unroll
  for (int v = 0; v < 8; ++v) {
    const int i = i0 + v + 8 * hi;  // C/D layout: VGPR v holds row v + 8*hi
    sqi[v] = sq[i];
    ti[v] = tgt[i];
  }

  float pbest[8], nbest[8];
  int pidx[8], nidx[8];
#pragma unroll
  for (int v = 0; v < 8; ++v) {
    pbest[v] = -__builtin_inff();
    nbest[v] = __builtin_inff();
    pidx[v] = 0x7fffffff;
    nidx[v] = 0x7fffffff;
  }

  for (int t = wave; t < NTILE; t += 8) {  // wave-uniform loop (EXEC stays all-1s)
    const int j0 = t * 16;
    const float* brow = X + (size_t)(j0 + lh) * DD + 2 * hi;
    if (t + 8 < NTILE)  // speculative prefetch of next B strip -> global_prefetch_b8
      __builtin_prefetch(X + (size_t)(j0 + 128 + lh) * DD, 0, 0);
    const float sqj = sq[j0 + lh];
    const long long tj = tgt[j0 + lh];

    v8f c = {};
#pragma unroll
    for (int s = 0; s < 32; ++s) {  // K = 128 = 32 x (16x16x4) f32 WMMAs
      v2f b = *(const v2f*)(brow + 4 * s);
      c = __builtin_amdgcn_wmma_f32_16x16x4_f32(false, a[s], false, b,
                                                (short)0, c, false, false);
    }

    const int j = j0 + lh;
#pragma unroll
    for (int v = 0; v < 8; ++v) {
      const int i = i0 + v + 8 * hi;
      float d2 = sqi[v] + sqj - 2.0f * c[v];
      d2 = fminf(fmaxf(d2, 1e-12f), 1e12f);
      const float d = sqrtf(d2);
      __builtin_nontemporal_store(d, &dist[(size_t)i * NN + j]);
      const bool same = (ti[v] == tj);
      const float pc = same ? d : -__builtin_inff();
      const float nc = same ? __builtin_inff() : d;
      if (pc > pbest[v]) { pbest[v] = pc; pidx[v] = j; }
      if (nc < nbest[v]) { nbest[v] = nc; nidx[v] = j; }
    }
  }

  // Reduce across the 16 lanes of each half-wave (rows are per (v,hi)).
#pragma unroll
  for (int v = 0; v < 8; ++v) {
    float pv = pbest[v]; int pj = pidx[v];
    float nv = nbest[v]; int nj = nidx[v];
#pragma unroll
    for (int off = 8; off > 0; off >>= 1) {
      float opv = __shfl_xor(pv, off, 16);
      int opj = __shfl_xor(pj, off, 16);
      float onv = __shfl_xor(nv, off, 16);
      int onj = __shfl_xor(nj, off, 16);
      if (opv > pv || (opv == pv && opj < pj)) { pv = opv; pj = opj; }
      if (onv < nv || (onv == nv && onj < nj)) { nv = onv; nj = onj; }
    }
    if (lh == 0) {
      const int r = v + 8 * hi;
      lds_pv[wave][r] = pv; lds_pj[wave][r] = pj;
      lds_nv[wave][r] = nv; lds_nj[wave][r] = nj;
    }
  }
  __syncthreads();

  // Reduce across the 8 waves; 16 threads write the 16 rows of this block.
  if (threadIdx.x < 16) {
    const int r = threadIdx.x;
    float pv = lds_pv[0][r]; int pj = lds_pj[0][r];
    float nv = lds_nv[0][r]; int nj = lds_nj[0][r];
#pragma unroll
    for (int w = 1; w < 8; ++w) {
      float opv = lds_pv[w][r]; int opj = lds_pj[w][r];
      float onv = lds_nv[w][r]; int onj = lds_nj[w][r];
      if (opv > pv || (opv == pv && opj < pj)) { pv = opv; pj = opj; }
      if (onv < nv || (onv == nv && onj < nj)) { nv = onv; nj = onj; }
    }
    dist_ap[i0 + r] = pv; p2[i0 + r] = pj;
    dist_an[i0 + r] = nv; n1[i0 + r] = nj;
  }
}

// ---------------------------------------------------------------------------
// Kernel 3: secondary-neighbor gathers + hinge means + precision.
// ---------------------------------------------------------------------------
__global__ __launch_bounds__(1024) void loss_kernel(
    const float* __restrict__ ap, const float* __restrict__ an,
    const int* __restrict__ p2, const int* __restrict__ n1,
    float* __restrict__ out) {
  __shared__ float acc[4];
  if (threadIdx.x < 4) acc[threadIdx.x] = 0.0f;
  __syncthreads();
  float s1 = 0.f, s2 = 0.f, s3 = 0.f, sp = 0.f;
  for (int i = threadIdx.x; i < NN; i += 1024) {
    const float dap = ap[i], dan = an[i];
    const float dn12 = an[n1[i]];  // hardest-neg of hardest-neg
    const float dp12 = ap[p2[i]];  // hardest-pos of hardest-pos
    s1 += fmaxf(dap - dan, 0.0f);   // margin-rank(dist_an, dist_ap), m=0
    s2 += fmaxf(dap - dn12, 0.0f);  // margin-rank(dist_n12, dist_ap)
    s3 += fmaxf(dp12 - dan, 0.0f);  // margin-rank(dist_an, dist_p12)
    sp += (dan > dap) ? 1.0f : 0.0f;
  }
#pragma unroll
  for (int off = 16; off > 0; off >>= 1) {
    s1 += __shfl_xor(s1, off, 32);
    s2 += __shfl_xor(s2, off, 32);
    s3 += __shfl_xor(s3, off, 32);
    sp += __shfl_xor(sp, off, 32);
  }
  if ((threadIdx.x & 31) == 0) {
    atomicAdd(&acc[0], s1);
    atomicAdd(&acc[1], s2);
    atomicAdd(&acc[2], s3);
    atomicAdd(&acc[3], sp);
  }
  __syncthreads();
  if (threadIdx.x == 0) {
    const float inv = 1.0f / (float)NN;
    out[0] = acc[0] * inv + 0.1f * (acc[1] * inv + acc[2] * inv);
    out[1] = acc[3] * inv;
  }
}

// ---------------------------------------------------------------------------
extern "C" void kernel_launch(void* const* d_in, const int* in_sizes, int n_in,
                              void* d_out, int out_size, void* d_ws, size_t ws_size,
                              hipStream_t stream) {
  const float* X = (const float*)d_in[0];
  const long long* tgt = (const long long*)d_in[1];
  // d_in[2] (cids) is unused by the reference computation.
  (void)in_sizes; (void)n_in; (void)out_size; (void)ws_size;

  float* out = (float*)d_out;
  float* dist = out + 2;  // (loss, prec, dist) concatenated flat

  float* sq = (float*)d_ws;
  float* dist_ap = sq + NN;
  float* dist_an = dist_ap + NN;
  int* p2 = (int*)(dist_an + NN);
  int* n1 = p2 + NN;

  sqnorm_kernel<<<NN / 8, 256, 0, stream>>>(X, sq);
  dist_tile_kernel<<<NN / 16, 256, 0, stream>>>(X, tgt, sq, dist,
                                                dist_ap, dist_an, p2, n1);
  loss_kernel<<<1, 1024, 0, stream>>>(dist_ap, dist_an, p2, n1, out);
}